// DNN_35064113004704
// MI455X (gfx1250) — compile-verified
//
#include <hip/hip_runtime.h>
#include <math.h>

#define WIN 320
#define HOP 160
#define NC  100
#define BATCH 16
#define LEN 160000
#define NT  999              // frames per batch item
#define MROWS (BATCH * NT)   // 15984 rows total
#define MTILES (MROWS / 16)  // 999 tiles of 16 rows
#define DSTRIDE 112          // padded column count of stored DCT slice

typedef __attribute__((ext_vector_type(2))) float v2f;
typedef __attribute__((ext_vector_type(8))) float v8f;

// D = A(16x4 f32) * B(4x16 f32) + C(16x16 f32), wave32 WMMA
__device__ __forceinline__ v8f wmma_f32_k4(v2f a, v2f b, v8f c) {
  return __builtin_amdgcn_wmma_f32_16x16x4_f32(false, a, false, b, (short)0, c,
                                               false, false);
}

// ---------------------------------------------------------------------------
// Build DCT matrix slice Dc[j][k] = D[j][k] for k<100 (0 padded to 112) and
// the periodic Hann window.
// ---------------------------------------------------------------------------
__global__ void init_const_kernel(float* __restrict__ Dc,
                                  float* __restrict__ win) {
  int i = blockIdx.x * blockDim.x + threadIdx.x;
  const float pi = 3.14159265358979323846f;
  if (i < WIN * DSTRIDE) {
    int n = i / DSTRIDE;
    int k = i % DSTRIDE;
    float v = 0.0f;
    if (k < NC) {
      v = sqrtf(2.0f / (float)WIN) *
          cosf((n + 0.5f) * pi / (float)WIN * (float)k);
      if (k == 0) v *= sqrtf(0.5f);
    }
    Dc[i] = v;
  } else if (i < WIN * DSTRIDE + WIN) {
    int j = i - WIN * DSTRIDE;
    win[j] = 0.5f * (1.0f - cosf(2.0f * pi * (float)j / (float)WIN));
  }
}

// ---------------------------------------------------------------------------
// Frame + window + forward DCT (clip to [-1,1], keep first 100 coeffs).
// grid = (999 M-tiles, 2 signals), block = 224 (7 waves; one N-tile per wave).
// ---------------------------------------------------------------------------
__global__ __launch_bounds__(224) void dct_kernel(
    const float* __restrict__ noisy, const float* __restrict__ clean,
    const float* __restrict__ Dc, const float* __restrict__ win,
    float* __restrict__ ndct, float* __restrict__ cdct) {
  __shared__ float A[16][WIN];  // 20 KB: 16 windowed frames
  const int mt = blockIdx.x;
  const int sid = blockIdx.y;
  const float* sig = (sid == 0) ? noisy : clean;
  float* dst = (sid == 0) ? ndct : cdct;
  const int tid = threadIdx.x;

  for (int e = tid; e < 16 * WIN; e += 224) {
    int r = e / WIN, j = e % WIN;
    int m = mt * 16 + r;
    int b = m / NT, t = m % NT;
    A[r][j] = sig[b * LEN + t * HOP + j] * win[j];
  }
  __syncthreads();

  const int wave = tid >> 5;
  const int lane = tid & 31;
  const int half = lane >> 4;
  const int l16  = lane & 15;
  const int n    = wave * 16 + l16;  // output coefficient index (0..111)

  v8f acc = {};
  for (int kb = 0; kb < WIN; kb += 4) {
    int ka = kb + 2 * half;
    v2f a  = {A[l16][ka], A[l16][ka + 1]};
    v2f bb = {Dc[ka * DSTRIDE + n], Dc[(ka + 1) * DSTRIDE + n]};
    acc = wmma_f32_k4(a, bb, acc);
  }

  if (n < NC) {
#pragma unroll
    for (int g = 0; g < 8; ++g) {
      int m = mt * 16 + g + 8 * half;
      float v = acc[g];
      v = fminf(1.0f, fmaxf(-1.0f, v));
      dst[m * NC + n] = v;
    }
  }
}

// ---------------------------------------------------------------------------
// 3-layer MLP (PReLU, PReLU, tanh) on 100-d coefficient vectors.
// One block (7 waves) owns a 16-row tile; activations ping-pong through LDS.
// ---------------------------------------------------------------------------
__global__ __launch_bounds__(224) void mlp_kernel(
    const float* __restrict__ xin, const float* __restrict__ W1,
    const float* __restrict__ b1, const float* __restrict__ p1,
    const float* __restrict__ W2, const float* __restrict__ b2,
    const float* __restrict__ p2, const float* __restrict__ W3,
    const float* __restrict__ b3, float* __restrict__ outdct) {
  __shared__ float X[16][NC];
  __shared__ float H[16][NC];
  const int mt = blockIdx.x;
  const int tid = threadIdx.x;

  for (int e = tid; e < 16 * NC; e += 224) {
    int r = e / NC, k = e % NC;
    X[r][k] = xin[(mt * 16 + r) * NC + k];
  }
  __syncthreads();

  const int wave = tid >> 5;
  const int lane = tid & 31;
  const int half = lane >> 4;
  const int l16  = lane & 15;
  const int n    = wave * 16 + l16;
  const int nb   = (n < NC) ? n : 0;  // clamp so weight loads stay in-bounds
  const float alpha1 = p1[0];
  const float alpha2 = p2[0];

  // ---- layer 1: h = prelu(X @ W1^T + b1) ----
  v8f acc = {};
  for (int kb = 0; kb < NC; kb += 4) {
    int ka = kb + 2 * half;
    v2f a  = {X[l16][ka], X[l16][ka + 1]};
    v2f bb = {W1[nb * NC + ka], W1[nb * NC + ka + 1]};
    acc = wmma_f32_k4(a, bb, acc);
  }
  if (n < NC) {
#pragma unroll
    for (int g = 0; g < 8; ++g) {
      float v = acc[g] + b1[n];
      H[g + 8 * half][n] = (v >= 0.0f) ? v : alpha1 * v;
    }
  }
  __syncthreads();

  // ---- layer 2: h = prelu(H @ W2^T + b2) ----
  acc = (v8f){};
  for (int kb = 0; kb < NC; kb += 4) {
    int ka = kb + 2 * half;
    v2f a  = {H[l16][ka], H[l16][ka + 1]};
    v2f bb = {W2[nb * NC + ka], W2[nb * NC + ka + 1]};
    acc = wmma_f32_k4(a, bb, acc);
  }
  if (n < NC) {
#pragma unroll
    for (int g = 0; g < 8; ++g) {
      float v = acc[g] + b2[n];
      X[g + 8 * half][n] = (v >= 0.0f) ? v : alpha2 * v;
    }
  }
  __syncthreads();

  // ---- layer 3: out = tanh(X @ W3^T + b3) ----
  acc = (v8f){};
  for (int kb = 0; kb < NC; kb += 4) {
    int ka = kb + 2 * half;
    v2f a  = {X[l16][ka], X[l16][ka + 1]};
    v2f bb = {W3[nb * NC + ka], W3[nb * NC + ka + 1]};
    acc = wmma_f32_k4(a, bb, acc);
  }
  if (n < NC) {
#pragma unroll
    for (int g = 0; g < 8; ++g) {
      int m = mt * 16 + g + 8 * half;
      outdct[m * NC + n] = tanhf(acc[g] + b3[n]);
    }
  }
}

// ---------------------------------------------------------------------------
// Inverse DCT: frames[m, j] = sum_k outdct[m,k] * D[j,k]   (j in 0..319)
// grid = (999, 4), block = 160 (5 waves; ntile = blockIdx.y*5 + wave).
// ---------------------------------------------------------------------------
__global__ __launch_bounds__(160) void idct_kernel(
    const float* __restrict__ outdct, const float* __restrict__ Dc,
    float* __restrict__ frames) {
  __shared__ float X[16][NC];
  const int mt = blockIdx.x;
  const int tid = threadIdx.x;

  for (int e = tid; e < 16 * NC; e += 160) {
    int r = e / NC, k = e % NC;
    X[r][k] = outdct[(mt * 16 + r) * NC + k];
  }
  __syncthreads();

  const int wave = tid >> 5;
  const int lane = tid & 31;
  const int half = lane >> 4;
  const int l16  = lane & 15;
  const int ntile = blockIdx.y * 5 + wave;  // 0..19
  const int n = ntile * 16 + l16;           // sample within frame, 0..319

  v8f acc = {};
  for (int kb = 0; kb < NC; kb += 4) {
    int ka = kb + 2 * half;
    v2f a  = {X[l16][ka], X[l16][ka + 1]};
    v2f bb = {Dc[n * DSTRIDE + ka], Dc[n * DSTRIDE + ka + 1]};
    acc = wmma_f32_k4(a, bb, acc);
  }
#pragma unroll
  for (int g = 0; g < 8; ++g) {
    int m = mt * 16 + g + 8 * half;
    frames[m * WIN + n] = acc[g];
  }
}

// ---------------------------------------------------------------------------
// Overlap-add, gather form (each sample has <= 2 contributing frames since
// HOP == WIN/2). Deterministic: no atomics.
// ---------------------------------------------------------------------------
__global__ void ola_kernel(const float* __restrict__ frames,
                           float* __restrict__ out) {
  int i = blockIdx.x * blockDim.x + threadIdx.x;
  if (i >= BATCH * LEN) return;
  int b = i / LEN;
  int n = i % LEN;
  int tq = n / HOP;
  float acc = 0.0f;
  if (tq < NT) acc += frames[(b * NT + tq) * WIN + (n - tq * HOP)];
  if (tq >= 1) acc += frames[(b * NT + tq - 1) * WIN + (n - (tq - 1) * HOP)];
  out[i] = acc;
}

// ---------------------------------------------------------------------------
extern "C" void kernel_launch(void* const* d_in, const int* in_sizes, int n_in,
                              void* d_out, int out_size, void* d_ws,
                              size_t ws_size, hipStream_t stream) {
  const float* noisy = (const float*)d_in[0];
  const float* clean = (const float*)d_in[1];
  const float* W1 = (const float*)d_in[2];
  const float* b1 = (const float*)d_in[3];
  const float* p1 = (const float*)d_in[4];
  const float* W2 = (const float*)d_in[5];
  const float* b2 = (const float*)d_in[6];
  const float* p2 = (const float*)d_in[7];
  const float* W3 = (const float*)d_in[8];
  const float* b3 = (const float*)d_in[9];

  float* out = (float*)d_out;
  float* out_dct   = out;                          // [M, 100]
  float* clean_dct = out + (size_t)MROWS * NC;     // [M, 100]
  float* speech    = out + (size_t)2 * MROWS * NC; // [B, L]

  float* wsf   = (float*)d_ws;
  float* Dc    = wsf;                               // 320*112
  float* win   = Dc + WIN * DSTRIDE;                // 320
  float* ndct  = win + WIN;                         // M*100
  float* fidct = ndct + (size_t)MROWS * NC;         // M*320

  init_const_kernel<<<(WIN * DSTRIDE + WIN + 255) / 256, 256, 0, stream>>>(Dc,
                                                                           win);
  dct_kernel<<<dim3(MTILES, 2), 224, 0, stream>>>(noisy, clean, Dc, win, ndct,
                                                  clean_dct);
  mlp_kernel<<<MTILES, 224, 0, stream>>>(ndct, W1, b1, p1, W2, b2, p2, W3, b3,
                                         out_dct);
  idct_kernel<<<dim3(MTILES, 4), 160, 0, stream>>>(out_dct, Dc, fidct);
  ola_kernel<<<(BATCH * LEN + 255) / 256, 256, 0, stream>>>(fidct, speech);
}